// QuanvolutionClassifier_65481071396423
// MI455X (gfx1250) — compile-verified
//
#include <hip/hip_runtime.h>
#include <hip/hip_bf16.h>
#include <math.h>

typedef __attribute__((ext_vector_type(2))) float v2f;
typedef __attribute__((ext_vector_type(8))) float v8f;

#define BSZ 1024
#define NPATCH 196          // 14*14 patches per image
#define FEATS 784           // 196*4 features per image
#define EMBED 128
#define NCLS 10

// ---------------------------------------------------------------------------
// Stage 1: quanvolution — one thread simulates one 4-qubit patch circuit.
// State = 16 complex amplitudes held entirely in registers (32 VGPRs).
// ---------------------------------------------------------------------------
__device__ __forceinline__ void apply1q(float re[16], float im[16], int mask,
                                        float u00r, float u00i, float u01r, float u01i,
                                        float u10r, float u10i, float u11r, float u11i) {
#pragma unroll
    for (int i = 0; i < 16; ++i) {
        if (i & mask) continue;
        const int j = i | mask;
        const float ar = re[i], ai = im[i];
        const float br = re[j], bi = im[j];
        re[i] = u00r * ar - u00i * ai + u01r * br - u01i * bi;
        im[i] = u00r * ai + u00i * ar + u01r * bi + u01i * br;
        re[j] = u10r * ar - u10i * ai + u11r * br - u11i * bi;
        im[j] = u10r * ai + u10i * ar + u11r * bi + u11i * br;
    }
}

__device__ __forceinline__ void cnot_gate(float re[16], float im[16], int cmask, int tmask) {
#pragma unroll
    for (int i = 0; i < 16; ++i) {
        if ((i & cmask) && !(i & tmask)) {
            const int j = i | tmask;
            float t = re[i]; re[i] = re[j]; re[j] = t;
            t = im[i]; im[i] = im[j]; im[j] = t;
        }
    }
}

__global__ void quanv_kernel(const float* __restrict__ x,
                             const float* __restrict__ qp,
                             float* __restrict__ feats) {
    const int t = blockIdx.x * blockDim.x + threadIdx.x;
    if (t >= BSZ * NPATCH) return;
    const int b  = t / NPATCH;
    const int pi = t - b * NPATCH;
    const int pr = pi / 14;
    const int pc = pi - pr * 14;

    const float* img = x + b * FEATS;   // 28x28 row-major
    float p0 = img[(2 * pr) * 28 + 2 * pc];
    float p1 = img[(2 * pr) * 28 + 2 * pc + 1];
    float p2 = img[(2 * pr + 1) * 28 + 2 * pc];
    float p3 = img[(2 * pr + 1) * 28 + 2 * pc + 1];

    // RY(p_w)|0> product-state encoder (real amplitudes)
    float q0c = cosf(0.5f * p0), q0s = sinf(0.5f * p0);
    float q1c = cosf(0.5f * p1), q1s = sinf(0.5f * p1);
    float q2c = cosf(0.5f * p2), q2s = sinf(0.5f * p2);
    float q3c = cosf(0.5f * p3), q3s = sinf(0.5f * p3);

    float re[16], im[16];
#pragma unroll
    for (int i = 0; i < 16; ++i) {
        const float a = (i & 8) ? q0s : q0c;
        const float bb = (i & 4) ? q1s : q1c;
        const float c = (i & 2) ? q2s : q2c;
        const float d = (i & 1) ? q3s : q3c;
        re[i] = a * bb * c * d;
        im[i] = 0.0f;
    }

    const float rh = 0.70710678118654752440f;
#pragma unroll
    for (int l = 0; l < 3; ++l) {
        const float t0 = qp[l * 3 + 0];
        const float t1 = qp[l * 3 + 1];
        const float t2 = qp[l * 3 + 2];
        const float c0 = cosf(0.5f * t0), s0 = sinf(0.5f * t0);
        const float c1 = cosf(0.5f * t1), s1 = sinf(0.5f * t1);
        const float c2 = cosf(0.5f * t2), s2 = sinf(0.5f * t2);
        // RX on wire 0 (bit mask 8)
        apply1q(re, im, 8, c0, 0.f, 0.f, -s0, 0.f, -s0, c0, 0.f);
        // RY on wire 1 (mask 4)
        apply1q(re, im, 4, c1, 0.f, -s1, 0.f, s1, 0.f, c1, 0.f);
        // RZ on wire 2 (mask 2): diag(e^{-it/2}, e^{+it/2})
        apply1q(re, im, 2, c2, -s2, 0.f, 0.f, 0.f, 0.f, c2, s2);
        // H on wire 3 (mask 1)
        apply1q(re, im, 1, rh, 0.f, rh, 0.f, rh, 0.f, -rh, 0.f);
        cnot_gate(re, im, 8, 4);
        cnot_gate(re, im, 4, 2);
        cnot_gate(re, im, 2, 1);
    }

    float z0 = 0.f, z1 = 0.f, z2 = 0.f, z3 = 0.f;
#pragma unroll
    for (int i = 0; i < 16; ++i) {
        const float pr2 = re[i] * re[i] + im[i] * im[i];
        z0 += (i & 8) ? -pr2 : pr2;
        z1 += (i & 4) ? -pr2 : pr2;
        z2 += (i & 2) ? -pr2 : pr2;
        z3 += (i & 1) ? -pr2 : pr2;
    }
    float* o = feats + b * FEATS + pi * 4;
    o[0] = z0; o[1] = z1; o[2] = z2; o[3] = z3;
}

// ---------------------------------------------------------------------------
// Stage 2: embed = relu(feats @ fc_w^T + fc_b), via V_WMMA_F32_16X16X4_F32.
// One wave per 16x16 output tile; 196 K-steps of K=4; two accumulators to
// break the WMMA->WMMA accumulation dependency.
// A (16x4 f32): lane m=lane&15 holds row m; lanes<16 carry K={0,1} in the
// two VGPRs, lanes>=16 carry K={2,3}. B (4x16) is symmetric with lane = N.
// C/D: VGPR r holds M=r (lanes 0-15) / M=r+8 (lanes 16-31), N = lane&15.
// ---------------------------------------------------------------------------
__global__ void fc_wmma_kernel(const float* __restrict__ feats,
                               const float* __restrict__ fc_w,
                               const float* __restrict__ fc_b,
                               float* __restrict__ embed) {
    const int wave  = blockIdx.x * (blockDim.x >> 5) + (threadIdx.x >> 5);
    const int tileN = wave & 7;          // 8 N-tiles (128/16)
    const int tileM = wave >> 3;         // 64 M-tiles (1024/16)
    const int lane  = threadIdx.x & 31;
    const int hi    = lane >> 4;         // 0: K={0,1}, 1: K={2,3}
    const int mn    = lane & 15;

    const float* Arow = feats + (tileM * 16 + mn) * FEATS;   // A row m
    const float* Brow = fc_w  + (tileN * 16 + mn) * FEATS;   // B col n = fc_w row n

    v8f acc0 = {};
    v8f acc1 = {};
    const int kofs = hi * 2;
#pragma unroll 4
    for (int kb = 0; kb < 196; kb += 2) {
        {
            const int k0 = kb * 4 + kofs;
            v2f a; a.x = Arow[k0]; a.y = Arow[k0 + 1];
            v2f bv; bv.x = Brow[k0]; bv.y = Brow[k0 + 1];
            acc0 = __builtin_amdgcn_wmma_f32_16x16x4_f32(
                false, a, false, bv, (short)0, acc0, false, false);
        }
        {
            const int k0 = (kb + 1) * 4 + kofs;
            v2f a; a.x = Arow[k0]; a.y = Arow[k0 + 1];
            v2f bv; bv.x = Brow[k0]; bv.y = Brow[k0 + 1];
            acc1 = __builtin_amdgcn_wmma_f32_16x16x4_f32(
                false, a, false, bv, (short)0, acc1, false, false);
        }
    }

    const float bias = fc_b[tileN * 16 + mn];
#pragma unroll
    for (int r = 0; r < 8; ++r) {
        const int M = tileM * 16 + r + hi * 8;
        float v = acc0[r] + acc1[r] + bias;
        v = v > 0.0f ? v : 0.0f;
        embed[M * EMBED + tileN * 16 + mn] = v;
    }
}

// ---------------------------------------------------------------------------
// Stage 3: logits = embed @ cls_w^T + cls_b; log_softmax per row.
// One wave per row: lanes 0..9 each own one class; wave32 shuffle reductions.
// ---------------------------------------------------------------------------
__global__ void head_kernel(const float* __restrict__ embed,
                            const float* __restrict__ cls_w,
                            const float* __restrict__ cls_b,
                            float* __restrict__ out) {
    const int row  = blockIdx.x * (blockDim.x >> 5) + (threadIdx.x >> 5);
    const int lane = threadIdx.x & 31;
    if (row >= BSZ) return;

    const float* e = embed + row * EMBED;
    float logit = -__builtin_inff();
    if (lane < NCLS) {
        float s = cls_b[lane];
        const float* w = cls_w + lane * EMBED;
#pragma unroll 4
        for (int k = 0; k < EMBED; ++k) s += e[k] * w[k];
        logit = s;
    }
    // wave-wide max (invalid lanes hold -inf)
    float m = logit;
    for (int off = 16; off > 0; off >>= 1)
        m = fmaxf(m, __shfl_xor(m, off, 32));
    float ex = (lane < NCLS) ? expf(logit - m) : 0.0f;
    float ssum = ex;
    for (int off = 16; off > 0; off >>= 1)
        ssum += __shfl_xor(ssum, off, 32);
    if (lane < NCLS)
        out[row * NCLS + lane] = logit - m - logf(ssum);
}

// ---------------------------------------------------------------------------
extern "C" void kernel_launch(void* const* d_in, const int* in_sizes, int n_in,
                              void* d_out, int out_size, void* d_ws, size_t ws_size,
                              hipStream_t stream) {
    const float* x     = (const float*)d_in[0];   // (1024, 784)
    const float* qp    = (const float*)d_in[1];   // (3, 3)
    const float* fc_w  = (const float*)d_in[2];   // (128, 784)
    const float* fc_b  = (const float*)d_in[3];   // (128,)
    const float* cls_w = (const float*)d_in[4];   // (10, 128)
    const float* cls_b = (const float*)d_in[5];   // (10,)

    float* out    = (float*)d_out;
    float* logits = out;                    // 1024*10 log_softmax
    float* embed  = out + BSZ * NCLS;       // 1024*128 embed (2nd tuple element)
    float* feats  = (float*)d_ws;           // 1024*784 scratch

    // 200704 patch threads
    quanv_kernel<<<(BSZ * NPATCH + 255) / 256, 256, 0, stream>>>(x, qp, feats);
    // 512 waves = 64 blocks x 8 waves; one 16x16 tile each
    fc_wmma_kernel<<<64, 256, 0, stream>>>(feats, fc_w, fc_b, embed);
    // 1024 rows, 8 rows (waves) per block
    head_kernel<<<BSZ / 8, 256, 0, stream>>>(embed, cls_w, cls_b, logits);
}